// GCN_interaction_dis_9268539425561
// MI455X (gfx1250) — compile-verified
//
#include <hip/hip_runtime.h>
#include <hip/hip_bf16.h>
#include <stdint.h>

// GCN scatter-add, fused: out[dst] += feature[src] * (ci[src]*mask/keep) * ci[dst]
// Memory-bound (atomic/L2) workload; WMMA inapplicable (no dense matmul).
// CDNA5 paths used: global_load_async_to_lds_b128 staging of edge streams,
// s_wait_asynccnt, LDS broadcast reads, no-return global_atomic_add_f32.

#define D_FEAT 32
#define EPB    256   // edges per block
#define BLK    256   // threads per block (8 waves of 32)

__device__ __forceinline__ uint32_t lds_off(const void* p) {
    // generic pointer to LDS: low 32 bits are the LDS byte offset (ISA 10.2)
    return (uint32_t)(uintptr_t)p;
}

__global__ void __launch_bounds__(BLK)
gcn_zero_kernel(float4* __restrict__ out, int n4) {
    int i = blockIdx.x * BLK + threadIdx.x;
    if (i < n4) out[i] = make_float4(0.f, 0.f, 0.f, 0.f);
}

__global__ void __launch_bounds__(BLK)
gcn_edge_kernel(const float* __restrict__ feature,
                const float* __restrict__ ci,
                const float* __restrict__ drop_rand,
                const int*   __restrict__ src,
                const int*   __restrict__ dst,
                float*       __restrict__ out,
                int E) {
    constexpr float KEEP = 1.0f - 0.7f;        // match reference's 1 - 0.7 in f32
    constexpr float INVK = 1.0f / KEEP;

    __shared__ int   s_src[EPB];
    __shared__ int   s_dst[EPB];
    __shared__ float s_dr [EPB];
    __shared__ float s_co [EPB];

    const int t  = threadIdx.x;
    const int e0 = blockIdx.x * EPB;

    // ---- Stage 1: async-copy edge metadata (src/dst/drop_rand) into LDS ----
    if (e0 + EPB <= E) {
        // full block: 3 arrays * 256 elems * 4B = 3072B = 192 x b128 chunks
        if (t < 192) {
            const int grp = t >> 6;      // 0=src, 1=dst, 2=drop_rand
            const int c   = t & 63;      // 16B chunk index (64 chunks per array)
            uint64_t gaddr;
            uint32_t laddr;
            if (grp == 0) {
                gaddr = (uint64_t)(uintptr_t)(src + e0) + (uint64_t)c * 16u;
                laddr = lds_off(s_src) + (uint32_t)c * 16u;
            } else if (grp == 1) {
                gaddr = (uint64_t)(uintptr_t)(dst + e0) + (uint64_t)c * 16u;
                laddr = lds_off(s_dst) + (uint32_t)c * 16u;
            } else {
                gaddr = (uint64_t)(uintptr_t)(drop_rand + e0) + (uint64_t)c * 16u;
                laddr = lds_off(s_dr) + (uint32_t)c * 16u;
            }
            // GV mode: LDS dest offset in one VGPR, 64-bit global addr in VGPR pair
            asm volatile("global_load_async_to_lds_b128 %0, %1, off"
                         :: "v"(laddr), "v"(gaddr) : "memory");
        }
        asm volatile("s_wait_asynccnt 0x0" ::: "memory");
    } else {
        // tail block: guarded scalar fill
        const int e = e0 + t;
        if (e < E) {
            s_src[t] = src[e];
            s_dst[t] = dst[e];
            s_dr[t]  = drop_rand[e];
        } else {
            s_dr[t] = 1.0f;              // force-drop out-of-range slots
        }
    }
    __syncthreads();

    // ---- Stage 2: one coefficient per edge (dropout fused with ci scaling) ----
    {
        float co = 0.0f;
        if (s_dr[t] < KEEP) {
            // fold ci[dst] in: sum(m)*ci[dst] == sum(m*ci[dst])
            co = ci[s_src[t]] * ci[s_dst[t]] * INVK;
        }
        s_co[t] = co;
    }
    __syncthreads();

    // ---- Stage 3: gather + scatter-add. 8 threads per edge, float4 each. ----
    const int sub = t & 7;               // which 16B slice of the 128B row
    #pragma unroll
    for (int i = 0; i < 8; ++i) {
        const int el = i * 32 + (t >> 3);        // edge slot in [0,256)
        const float coeff = s_co[el];
        if (coeff == 0.0f) continue;             // dropped edge (or OOB tail)
        const int s = s_src[el];
        const int d = s_dst[el];

        const float4 f = *reinterpret_cast<const float4*>(
            feature + (size_t)s * D_FEAT + (size_t)sub * 4);

        float* o = out + (size_t)d * D_FEAT + (size_t)sub * 4;
        // no-return fp32 global atomics (result discarded -> STOREcnt path)
        (void)__hip_atomic_fetch_add(o + 0, f.x * coeff, __ATOMIC_RELAXED, __HIP_MEMORY_SCOPE_AGENT);
        (void)__hip_atomic_fetch_add(o + 1, f.y * coeff, __ATOMIC_RELAXED, __HIP_MEMORY_SCOPE_AGENT);
        (void)__hip_atomic_fetch_add(o + 2, f.z * coeff, __ATOMIC_RELAXED, __HIP_MEMORY_SCOPE_AGENT);
        (void)__hip_atomic_fetch_add(o + 3, f.w * coeff, __ATOMIC_RELAXED, __HIP_MEMORY_SCOPE_AGENT);
    }
}

extern "C" void kernel_launch(void* const* d_in, const int* in_sizes, int n_in,
                              void* d_out, int out_size, void* d_ws, size_t ws_size,
                              hipStream_t stream) {
    const float* feature   = (const float*)d_in[0];
    const float* ci        = (const float*)d_in[1];
    const float* drop_rand = (const float*)d_in[2];
    const int*   src       = (const int*)d_in[3];
    const int*   dst       = (const int*)d_in[4];
    float*       out       = (float*)d_out;

    const int E  = in_sizes[3];
    const int n4 = out_size / 4;                 // out is N*32 floats, 16B aligned

    // zero-init accumulator output (harness poisons d_out)
    gcn_zero_kernel<<<(n4 + BLK - 1) / BLK, BLK, 0, stream>>>(
        reinterpret_cast<float4*>(out), n4);

    // fused dropout + gather + scatter-add + ci[dst] scaling
    const int blocks = (E + EPB - 1) / EPB;
    gcn_edge_kernel<<<blocks, BLK, 0, stream>>>(
        feature, ci, drop_rand, src, dst, out, E);
}